// CAM_Module_335007449966
// MI455X (gfx1250) — compile-verified
//
#include <hip/hip_runtime.h>

typedef __attribute__((ext_vector_type(2))) float v2f;
typedef __attribute__((ext_vector_type(8))) float v8f;

#define B_  16
#define C_  512
#define N_  4096
#define PAD_A 68
#define PAD_B 72

__device__ __forceinline__ v8f wmma4(v2f a, v2f b, v8f c) {
    // D = A(16x4) * B(4x16) + C  in fp32
    return __builtin_amdgcn_wmma_f32_16x16x4_f32(false, a, false, b, (short)0, c, false, false);
}

// ---------------------------------------------------------------------------
// Kernel 1: energy[b,i,j] = sum_n q[b,i,n]*q[b,j,n]   (symmetric!)
// Only upper-triangular 64x64 blocks (36 of 64); mirror-store off-diagonals.
// 128 threads = 4 waves; each wave owns a 32x32 quadrant (2x2 WMMA tiles).
// Double-buffered LDS, register-staged global loads.
// ---------------------------------------------------------------------------
__global__ __launch_bounds__(128) void cam_energy(const float* __restrict__ x,
                                                  float* __restrict__ energy) {
    __shared__ float As[2][64 * PAD_A];
    __shared__ float Bs[2][64 * PAD_A];

    const int b = blockIdx.y;
    // decode triangular block index (36 = 8+7+...+1)
    int rem = blockIdx.x, bi = 0;
    while (rem >= 8 - bi) { rem -= 8 - bi; ++bi; }
    const int bj = bi + rem;
    const int i0 = bi << 6, j0 = bj << 6;

    const int t    = threadIdx.x;
    const int lane = t & 31;
    const int w    = t >> 5;          // wave 0..3
    const int wi   = w & 1;           // 32-row half
    const int wj   = w >> 1;          // 32-col half
    const int m    = lane & 15;
    const int hh   = lane >> 4;

    const float* q  = x + (size_t)b * C_ * N_;
    const int r0 = t >> 4;            // 0..7
    const int cc = (t & 15) << 2;

    const float* gA = q + (size_t)(i0 + r0) * N_ + cc;
    const float* gB = q + (size_t)(j0 + r0) * N_ + cc;
    const int soff  = r0 * PAD_A + cc;

    v8f acc[2][2];
    #pragma unroll
    for (int ii = 0; ii < 2; ++ii)
        #pragma unroll
        for (int jj = 0; jj < 2; ++jj)
            acc[ii][jj] = (v8f){0.f,0.f,0.f,0.f,0.f,0.f,0.f,0.f};

    float4 ra[8], rb[8];
    #pragma unroll
    for (int i = 0; i < 8; ++i) {
        ra[i] = *(const float4*)(gA + (size_t)(8 * i) * N_);
        rb[i] = *(const float4*)(gB + (size_t)(8 * i) * N_);
    }
    #pragma unroll
    for (int i = 0; i < 8; ++i) {
        *(float4*)(&As[0][soff + i * 8 * PAD_A]) = ra[i];
        *(float4*)(&Bs[0][soff + i * 8 * PAD_A]) = rb[i];
    }
    __syncthreads();

    const int NC = N_ / 64;  // 64 chunks
    #pragma unroll 1
    for (int c = 0; c < NC; ++c) {
        const int p = c & 1;
        if (c + 1 < NC) {
            const int k0 = (c + 1) * 64;
            #pragma unroll
            for (int i = 0; i < 8; ++i) {
                ra[i] = *(const float4*)(gA + (size_t)(8 * i) * N_ + k0);
                rb[i] = *(const float4*)(gB + (size_t)(8 * i) * N_ + k0);
            }
        }
        const float* Ab = &As[p][(wi * 32 + m) * PAD_A];
        const float* Bb = &Bs[p][(wj * 32 + m) * PAD_A];
        #pragma unroll
        for (int kk = 0; kk < 64; kk += 4) {
            const int klo = kk + hh * 2;
            v2f a0 = *(const v2f*)(Ab + klo);
            v2f a1 = *(const v2f*)(Ab + 16 * PAD_A + klo);
            v2f b0 = *(const v2f*)(Bb + klo);
            v2f b1 = *(const v2f*)(Bb + 16 * PAD_A + klo);
            acc[0][0] = wmma4(a0, b0, acc[0][0]);
            acc[0][1] = wmma4(a0, b1, acc[0][1]);
            acc[1][0] = wmma4(a1, b0, acc[1][0]);
            acc[1][1] = wmma4(a1, b1, acc[1][1]);
        }
        if (c + 1 < NC) {
            const int pn = p ^ 1;
            #pragma unroll
            for (int i = 0; i < 8; ++i) {
                *(float4*)(&As[pn][soff + i * 8 * PAD_A]) = ra[i];
                *(float4*)(&Bs[pn][soff + i * 8 * PAD_A]) = rb[i];
            }
        }
        __syncthreads();
    }

    float* e = energy + (size_t)b * C_ * C_;
    const bool mirror = (bi != bj);
    #pragma unroll
    for (int ii = 0; ii < 2; ++ii)
        #pragma unroll
        for (int jj = 0; jj < 2; ++jj)
            #pragma unroll
            for (int v = 0; v < 8; ++v) {
                const int row = i0 + wi * 32 + ii * 16 + v + hh * 8;
                const int col = j0 + wj * 32 + jj * 16 + m;
                const float val = acc[ii][jj][v];
                e[(size_t)row * C_ + col] = val;
                if (mirror) e[(size_t)col * C_ + row] = val;
            }
}

// ---------------------------------------------------------------------------
// Kernel 2: attention = softmax(rowmax(e) - e) == exp(rowmin - e)/sum (exact)
// One 256-thread block per row of 512.
// ---------------------------------------------------------------------------
__global__ __launch_bounds__(256) void cam_softmax(float* __restrict__ energy) {
    __shared__ float red[256];
    const int t = threadIdx.x;
    float* e = energy + (size_t)blockIdx.x * C_;

    float a0 = e[t];
    float a1 = e[t + 256];

    red[t] = fminf(a0, a1);
    __syncthreads();
    #pragma unroll
    for (int s = 128; s > 0; s >>= 1) {
        if (t < s) red[t] = fminf(red[t], red[t + s]);
        __syncthreads();
    }
    float mn = red[0];
    __syncthreads();

    float p0 = __expf(mn - a0);
    float p1 = __expf(mn - a1);

    red[t] = p0 + p1;
    __syncthreads();
    #pragma unroll
    for (int s = 128; s > 0; s >>= 1) {
        if (t < s) red[t] += red[t + s];
        __syncthreads();
    }
    float inv = 1.0f / red[0];

    e[t]       = p0 * inv;
    e[t + 256] = p1 * inv;
}

// ---------------------------------------------------------------------------
// Kernel 3: out[b,i,n] = gamma * sum_j att[b,i,j] * q[b,j,n] + x[b,i,n]
// 64(i) x 64(n) block tile, K=j=512 in 8 chunks; same 2x2 wave micro-tile,
// double-buffered LDS. B operand read column-wise from pad-72 q tile.
// ---------------------------------------------------------------------------
__global__ __launch_bounds__(128) void cam_out(const float* __restrict__ x,
                                               const float* __restrict__ att,
                                               const float* __restrict__ gamma_p,
                                               float* __restrict__ out) {
    __shared__ float As[2][64 * PAD_A];   // att tile [i][j]
    __shared__ float Bs[2][64 * PAD_B];   // q   tile [j][n]

    const int b  = blockIdx.y;
    const int i0 = (blockIdx.x >> 6) << 6;   // 8 i-blocks
    const int n0 = (blockIdx.x & 63) << 6;   // 64 n-blocks

    const int t    = threadIdx.x;
    const int lane = t & 31;
    const int w    = t >> 5;
    const int wi   = w & 1;
    const int wj   = w >> 1;
    const int m    = lane & 15;
    const int hh   = lane >> 4;

    const float* q = x   + (size_t)b * C_ * N_;
    const float* A = att + (size_t)b * C_ * C_;
    const int r0 = t >> 4;
    const int cc = (t & 15) << 2;

    const float* gA = A + (size_t)(i0 + r0) * C_ + cc;     // + j0 per chunk
    const float* gB = q + (size_t)r0 * N_ + n0 + cc;       // + j0*N per chunk
    const int soffA = r0 * PAD_A + cc;
    const int soffB = r0 * PAD_B + cc;

    v8f acc[2][2];
    #pragma unroll
    for (int ii = 0; ii < 2; ++ii)
        #pragma unroll
        for (int jj = 0; jj < 2; ++jj)
            acc[ii][jj] = (v8f){0.f,0.f,0.f,0.f,0.f,0.f,0.f,0.f};

    float4 ra[8], rb[8];
    #pragma unroll
    for (int i = 0; i < 8; ++i) {
        ra[i] = *(const float4*)(gA + (size_t)(8 * i) * C_);
        rb[i] = *(const float4*)(gB + (size_t)(8 * i) * N_);
    }
    #pragma unroll
    for (int i = 0; i < 8; ++i) {
        *(float4*)(&As[0][soffA + i * 8 * PAD_A]) = ra[i];
        *(float4*)(&Bs[0][soffB + i * 8 * PAD_B]) = rb[i];
    }
    __syncthreads();

    const int NC = C_ / 64;  // 8 chunks
    #pragma unroll 1
    for (int c = 0; c < NC; ++c) {
        const int p = c & 1;
        if (c + 1 < NC) {
            const int j0n = (c + 1) * 64;
            #pragma unroll
            for (int i = 0; i < 8; ++i) {
                ra[i] = *(const float4*)(gA + (size_t)(8 * i) * C_ + j0n);
                rb[i] = *(const float4*)(gB + (size_t)(j0n + 8 * i) * N_);
            }
        }
        const float* Ab = &As[p][(wi * 32 + m) * PAD_A];
        const float* Bb = &Bs[p][0];
        const int tc0 = wj * 32 + m;
        const int tc1 = tc0 + 16;
        #pragma unroll
        for (int kk = 0; kk < 64; kk += 4) {
            const int klo = kk + hh * 2;
            v2f a0 = *(const v2f*)(Ab + klo);
            v2f a1 = *(const v2f*)(Ab + 16 * PAD_A + klo);
            v2f b0, b1;
            b0.x = Bb[(klo + 0) * PAD_B + tc0];
            b0.y = Bb[(klo + 1) * PAD_B + tc0];
            b1.x = Bb[(klo + 0) * PAD_B + tc1];
            b1.y = Bb[(klo + 1) * PAD_B + tc1];
            acc[0][0] = wmma4(a0, b0, acc[0][0]);
            acc[0][1] = wmma4(a0, b1, acc[0][1]);
            acc[1][0] = wmma4(a1, b0, acc[1][0]);
            acc[1][1] = wmma4(a1, b1, acc[1][1]);
        }
        if (c + 1 < NC) {
            const int pn = p ^ 1;
            #pragma unroll
            for (int i = 0; i < 8; ++i) {
                *(float4*)(&As[pn][soffA + i * 8 * PAD_A]) = ra[i];
                *(float4*)(&Bs[pn][soffB + i * 8 * PAD_B]) = rb[i];
            }
        }
        __syncthreads();
    }

    const float g = gamma_p[0];
    const float* xb = q;                         // x, same batch base
    float*       ob = out + (size_t)b * C_ * N_;
    #pragma unroll
    for (int ii = 0; ii < 2; ++ii)
        #pragma unroll
        for (int jj = 0; jj < 2; ++jj)
            #pragma unroll
            for (int v = 0; v < 8; ++v) {
                const int row = i0 + wi * 32 + ii * 16 + v + hh * 8;
                const int col = n0 + wj * 32 + jj * 16 + m;
                const size_t o = (size_t)row * N_ + col;
                ob[o] = fmaf(g, acc[ii][jj][v], xb[o]);
            }
}

// ---------------------------------------------------------------------------
extern "C" void kernel_launch(void* const* d_in, const int* in_sizes, int n_in,
                              void* d_out, int out_size, void* d_ws, size_t ws_size,
                              hipStream_t stream) {
    const float* x     = (const float*)d_in[0];
    const float* gamma = (const float*)d_in[1];
    float* energy = (float*)d_ws;          // B*C*C floats = 16 MiB scratch
    float* out    = (float*)d_out;

    cam_energy <<<dim3(36, B_),  dim3(128), 0, stream>>>(x, energy);
    cam_softmax<<<dim3(B_ * C_), dim3(256), 0, stream>>>(energy);
    cam_out    <<<dim3(512, B_), dim3(128), 0, stream>>>(x, energy, gamma, out);
}